// PrimaryNetwork_67018669686909
// MI455X (gfx1250) — compile-verified
//
#include <hip/hip_runtime.h>
#include <hip/hip_bf16.h>

typedef __attribute__((ext_vector_type(2))) float v2f;
typedef __attribute__((ext_vector_type(4))) float v4f;
typedef __attribute__((ext_vector_type(8))) float v8f;

// ---------------- constant tables (ZS_SIZE / OFFS from reference) ----------
__constant__ int c_OFFS[37] = {
  0,1,2,3,4,5,6,7,8,9,10,11,12,14,18,22,26,30,34,38,42,46,50,54,58,66,
  82,98,114,130,146,162,178,194,210,226,242};
__constant__ int c_ZSK[36] = {
  1,1,1,1,1,1,1,1,1,1,1,1,1, 2,2,2,2,2,2,2,2,2,2,2,2, 4,4,4,4,4,4,4,4,4,4,4};

static const int h_OFFS[37] = {
  0,1,2,3,4,5,6,7,8,9,10,11,12,14,18,22,26,30,34,38,42,46,50,54,58,66,
  82,98,114,130,146,162,178,194,210,226,242};
static const int h_ZSH[36] = {
  1,1,1,1,1,1,1,1,1,1,1,1, 2, 2,2,2,2,2,2,2,2,2,2,2, 4, 4,4,4,4,4,4,4,4,4,4,4};
static const int h_ZSK[36] = {
  1,1,1,1,1,1,1,1,1,1,1,1,1, 2,2,2,2,2,2,2,2,2,2,2,2, 4,4,4,4,4,4,4,4,4,4,4};

// Pair-interleaved weight layout inside one [Kpad][16] m-tile:
//   element (kk, m) -> (kk>>2)*64 + ((kk&3)>>1)*32 + m*2 + (kk&1)
// so the A fragment (K=kA, kA+1 at row m) is one contiguous v2f.
__device__ __forceinline__ long woff(long mt, int Kpad, int kk, int m)
{
    int q = kk >> 2, j = kk & 3;
    return mt * Kpad * 16 + q * 64 + ((j >> 1) << 5) + (m << 1) + (j & 1);
}

// ---------------- hypernetwork weight generation ---------------------------
// One block (256 threads) per z-row r. K ordering = (tap, ic): kk = tap*Cin + ic.
__global__ void hyper_gen(const float* __restrict__ z_all,
                          const float* __restrict__ w2, const float* __restrict__ b2,
                          const float* __restrict__ w1, const float* __restrict__ b1,
                          float* __restrict__ wk)
{
    __shared__ float zsh[64];
    __shared__ float hin[1024];
    const int r = blockIdx.x;
    const int t = threadIdx.x;

    if (t < 64) zsh[t] = z_all[r * 64 + t];
    __syncthreads();

    for (int j = t; j < 1024; j += 256) {
        float s = b2[j];
        #pragma unroll 8
        for (int d = 0; d < 64; ++d) s += zsh[d] * w2[d * 1024 + j];
        hin[j] = s;
    }
    __syncthreads();

    // locate layer / (hi, ki)
    int l = 0;
    while (c_OFFS[l + 1] <= r) ++l;
    const int rr = r - c_OFFS[l];
    const int kk_ = c_ZSK[l];
    const int hi = rr / kk_;
    const int ki = rr % kk_;
    const int Cin = kk_ * 16;
    const int Kpad = 9 * Cin;                 // == kk_ * 144
    float* wl = wk + (long)c_OFFS[l] * 2304;  // layer base

    // 2304 outputs (io:16 x o:144), 9 per thread
    for (int e = 0; e < 9; ++e) {
        const int idx = t * 9 + e;
        const int io  = idx / 144;
        const int o   = idx - io * 144;
        float s = b1[o];
        #pragma unroll 8
        for (int d2 = 0; d2 < 64; ++d2) s += hin[io * 64 + d2] * w1[d2 * 144 + o];
        const int ii  = o / 9;
        const int tap = o - ii * 9;
        const int kkid = tap * Cin + ki * 16 + ii;
        wl[woff(hi, Kpad, kkid, io)] = s;
    }
}

// conv1_w (16,3,3,3) -> Cpad=4 layout: kk = tap*4 + ic, Kpad = 36.
__global__ void repack_conv1(const float* __restrict__ src, float* __restrict__ dst)
{
    int t = blockIdx.x * blockDim.x + threadIdx.x;   // 16*36 = 576 elements
    if (t >= 576) return;
    int oc = t / 36;
    int kk = t - oc * 36;
    int tap = kk >> 2;
    int ic  = kk & 3;
    float v = (ic < 3) ? src[oc * 27 + ic * 9 + tap] : 0.0f;
    dst[woff(0, 36, kk, oc)] = v;
}

// res 1x1 conv (Cout,Cin,1,1) -> [mt][Cin][16] paired layout, kk = ic.
__global__ void repack_res(const float* __restrict__ src, float* __restrict__ dst,
                           int Cout, int Cin)
{
    int t = blockIdx.x * blockDim.x + threadIdx.x;
    if (t >= Cout * Cin) return;
    int oc = t / Cin;
    int ic = t - oc * Cin;
    dst[woff(oc >> 4, Cin, ic, oc & 15)] = src[oc * Cin + ic];
}

// x (1024,3,32,32) NCHW -> X4 (1024,32,32,4) NHWC, channel 3 zeroed.
__global__ void to_nhwc4(const float* __restrict__ src, float* __restrict__ dst)
{
    int t = blockIdx.x * blockDim.x + threadIdx.x;   // over 1024*1024 pixels
    if (t >= 1024 * 1024) return;
    int pixel = t & 1023;        // y*32+x
    int n = t >> 10;
    const float* s = src + (long)n * 3 * 1024 + pixel;
    float* d = dst + (long)t * 4;
    d[0] = s[0];
    d[1] = s[1024];
    d[2] = s[2048];
    d[3] = 0.0f;
}

// ---------------- implicit-GEMM conv via V_WMMA_F32_16X16X4_F32 ------------
// NHWC activations, compile-time channel count CP. One wave computes a
// 16(M) x 16(N) output tile for all MT m-tiles (B-fragment reuse).
// Block = (32, 8) = 8 waves. K ordering kk = tap*CP + ic: tap loop unrolled,
// bounds check hoisted per tap; channel loop has constant trip count with
// immediate-offset v2f loads feeding the WMMA chain.
template <int MT, int KS, int CP>
__global__ void conv_wmma(const float* __restrict__ in, const float* __restrict__ wk,
                          const float* __restrict__ bias,
                          const float* __restrict__ bng, const float* __restrict__ bnb,
                          const float* __restrict__ bnm, const float* __restrict__ bnv,
                          int bnrow, const float* res, float* out,
                          int NT, int Hin, int Win, int wsh, int hsh,
                          int stride, int relu)
{
    const int lane = threadIdx.x;                 // 0..31
    const int nt = blockIdx.x * 8 + threadIdx.y;
    if (nt >= NT) return;

    const int col  = lane & 15;                   // N within tile
    const int pix  = nt * 16 + col;
    const int ox   = pix & ((1 << wsh) - 1);
    const int oy   = (pix >> wsh) & ((1 << hsh) - 1);
    const int bimg = pix >> (wsh + hsh);
    const int half  = lane >> 4;                  // 0 / 1
    const int khalf = half << 1;                  // 0 / 2 : K sub-pair per lane half
    const int mrow  = lane & 15;                  // M row for A fragment
    constexpr int Kpad = (KS == 3) ? 9 * CP : CP;

    v8f acc[MT];
    #pragma unroll
    for (int m = 0; m < MT; ++m) acc[m] = (v8f){0, 0, 0, 0, 0, 0, 0, 0};

    const float* wbase = wk + (half << 5) + (mrow << 1);
    constexpr int ntaps = (KS == 3) ? 9 : 1;
    #pragma unroll
    for (int tap = 0; tap < ntaps; ++tap) {
        int iy, ix;
        bool ok;
        if (KS == 1) {
            iy = oy * stride; ix = ox * stride; ok = true;
        } else {
            const int kh = tap / 3, kw = tap - kh * 3;
            iy = oy * stride + kh - 1;
            ix = ox * stride + kw - 1;
            ok = ((unsigned)iy < (unsigned)Hin) & ((unsigned)ix < (unsigned)Win);
        }
        const float* bp = in + (((long)bimg * Hin + iy) * Win + ix) * CP + khalf;
        const float* ap = wbase + tap * CP * 16;  // k0 = tap*CP, paired layout

        #pragma unroll 4
        for (int icc = 0; icc < CP; icc += 4) {
            v2f bfrag;
            if (ok) bfrag = *(const v2f*)(bp + icc);
            else    bfrag = (v2f){0.0f, 0.0f};
            #pragma unroll
            for (int m = 0; m < MT; ++m) {
                v2f afrag = *(const v2f*)(ap + m * Kpad * 16 + (icc << 4));
                acc[m] = __builtin_amdgcn_wmma_f32_16x16x4_f32(
                    false, afrag, false, bfrag, (short)0, acc[m], false, false);
            }
        }
    }

    // ---- epilogue: NHWC, 8 consecutive channels per lane -> 2x b128 -------
    constexpr int Cout = MT * 16;
    const int half8 = half << 3;
    #pragma unroll
    for (int m = 0; m < MT; ++m) {
        float vals[8];
        const int ocb = m * 16 + half8;
        #pragma unroll
        for (int r = 0; r < 8; ++r) {
            const int oc = ocb + r;
            float v = acc[m][r];
            if (bias) v += bias[oc];
            if (bnrow >= 0) {
                const float g  = bng[bnrow * 64 + oc];
                const float be = bnb[bnrow * 64 + oc];
                const float mu = bnm[bnrow * 64 + oc];
                const float va = bnv[bnrow * 64 + oc];
                v = (v - mu) * (g * __frsqrt_rn(va + 1e-5f)) + be;
            }
            vals[r] = v;
        }
        const long obase = (long)pix * Cout + ocb;
        if (res) {
            v4f r0 = *(const v4f*)(res + obase);
            v4f r1 = *(const v4f*)(res + obase + 4);
            #pragma unroll
            for (int r = 0; r < 4; ++r) { vals[r] += r0[r]; vals[r + 4] += r1[r]; }
        }
        if (relu) {
            #pragma unroll
            for (int r = 0; r < 8; ++r) vals[r] = fmaxf(vals[r], 0.0f);
        }
        *(v4f*)(out + obase)     = (v4f){vals[0], vals[1], vals[2], vals[3]};
        *(v4f*)(out + obase + 4) = (v4f){vals[4], vals[5], vals[6], vals[7]};
    }
}

// ---------------- global average pool + final linear -----------------------
// Input NHWC (n, 8, 8, 64).
__global__ void pool_fc(const float* __restrict__ in, const float* __restrict__ fw,
                        const float* __restrict__ fb, float* __restrict__ out)
{
    __shared__ float avg[64];
    const int n = blockIdx.x;
    const int c = threadIdx.x;                    // 64 threads
    float s = 0.0f;
    const float* base = in + (long)n * 64 * 64 + c;
    #pragma unroll 8
    for (int p = 0; p < 64; ++p) s += base[p * 64];
    avg[c] = s * (1.0f / 64.0f);
    __syncthreads();
    if (c < 10) {
        float o = fb[c];
        #pragma unroll 8
        for (int d = 0; d < 64; ++d) o += avg[d] * fw[c * 64 + d];
        out[n * 10 + c] = o;
    }
}

// ---------------- host-side launch helpers ---------------------------------
static int ilog2i(int v) { int s = 0; while ((1 << s) < v) ++s; return s; }

static void launch_conv(const float* in, const float* wk, const float* bias,
                        const float* bng, const float* bnb, const float* bnm,
                        const float* bnv, int bnrow, const float* res, float* out,
                        int Bn, int Cpad, int Cout, int Hin, int Win,
                        int stride, int ksize, int relu, hipStream_t stream)
{
    const int pad  = (ksize == 3) ? 1 : 0;
    const int Hout = (Hin + 2 * pad - ksize) / stride + 1;
    const int Wout = (Win + 2 * pad - ksize) / stride + 1;
    const int NT = (Bn * Hout * Wout) / 16;
    const int wsh = ilog2i(Wout), hsh = ilog2i(Hout);
    dim3 blk(32, 8);
    dim3 grd((NT + 7) / 8);
    const int MT = Cout / 16;

#define CONV_CASE(MTV, KSV, CPV)                                                \
    conv_wmma<MTV, KSV, CPV><<<grd, blk, 0, stream>>>(                          \
        in, wk, bias, bng, bnb, bnm, bnv, bnrow, res, out, NT, Hin, Win,        \
        wsh, hsh, stride, relu)

    if (ksize == 3) {
        if (MT == 1 && Cpad == 4)        CONV_CASE(1, 3, 4);    // stem
        else if (MT == 1 && Cpad == 16)  CONV_CASE(1, 3, 16);   // stage1 16->16
        else if (MT == 2 && Cpad == 16)  CONV_CASE(2, 3, 16);   // block6 16->32
        else if (MT == 2 && Cpad == 32)  CONV_CASE(2, 3, 32);   // stage2 32->32
        else if (MT == 4 && Cpad == 32)  CONV_CASE(4, 3, 32);   // block12 32->64
        else                             CONV_CASE(4, 3, 64);   // stage3 64->64
    } else {
        if (MT == 2)                     CONV_CASE(2, 1, 16);   // res6 16->32
        else                             CONV_CASE(4, 1, 32);   // res12 32->64
    }
#undef CONV_CASE
}

extern "C" void kernel_launch(void* const* d_in, const int* in_sizes, int n_in,
                              void* d_out, int out_size, void* d_ws, size_t ws_size,
                              hipStream_t stream)
{
    (void)in_sizes; (void)n_in; (void)out_size; (void)ws_size;

    const float* x       = (const float*)d_in[0];
    const float* conv1_w = (const float*)d_in[1];
    const float* conv1_b = (const float*)d_in[2];
    const float* bng     = (const float*)d_in[3];
    const float* bnb     = (const float*)d_in[4];
    const float* bnm     = (const float*)d_in[5];
    const float* bnv     = (const float*)d_in[6];
    const float* hn_w2   = (const float*)d_in[7];
    const float* hn_b2   = (const float*)d_in[8];
    const float* hn_w1   = (const float*)d_in[9];
    const float* hn_b1   = (const float*)d_in[10];
    const float* z_all   = (const float*)d_in[11];
    const float* res_w6  = (const float*)d_in[12];
    const float* res_b6  = (const float*)d_in[13];
    const float* res_w12 = (const float*)d_in[14];
    const float* res_b12 = (const float*)d_in[15];
    const float* fin_w   = (const float*)d_in[16];
    const float* fin_b   = (const float*)d_in[17];
    float* out = (float*)d_out;

    // workspace layout (floats); all activations NHWC
    float* A    = (float*)d_ws;           // 1024*32*32*16 = 16777216
    float* Bb   = A    + 16777216;        // ping-pong buffer
    float* R    = Bb   + 16777216;        // residual (max 8388608); stem reuses as X4
    float* WKH  = R    + 8388608;         // hyper weights: 242*2304 = 557568
    float* WK1  = WKH  + 557568;          // conv1 weights: 36*16 = 576
    float* WR6  = WK1  + 576;             // res6 weights: 2*16*16
    float* WR12 = WR6  + 512;             // res12 weights: 4*32*16
    float* X4   = R;                      // stem NHWC input (1024*32*32*4), aliases R

    const int Bn = 1024;

    // 1) generate all hyper conv weights in WMMA layout
    hyper_gen<<<242, 256, 0, stream>>>(z_all, hn_w2, hn_b2, hn_w1, hn_b1, WKH);
    // 2) repack static weights + transpose stem input to NHWC(4)
    repack_conv1<<<3, 256, 0, stream>>>(conv1_w, WK1);
    repack_res<<<2, 256, 0, stream>>>(res_w6, WR6, 32, 16);
    repack_res<<<8, 256, 0, stream>>>(res_w12, WR12, 64, 32);
    to_nhwc4<<<4096, 256, 0, stream>>>(x, X4);

    // 3) stem: conv1 (4->16, 3x3, pad1) + bias + BN0 + ReLU -> A
    launch_conv(X4, WK1, conv1_b, bng, bnb, bnm, bnv, /*bnrow=*/0,
                nullptr, A, Bn, /*Cpad=*/4, /*Cout=*/16, 32, 32,
                /*stride=*/1, /*ksize=*/3, /*relu=*/1, stream);

    // 4) 18 residual blocks
    for (int i = 0; i < 18; ++i) {
        const int l1 = 2 * i, l2 = 2 * i + 1;
        const int h1 = h_ZSH[l1], k1 = h_ZSK[l1];
        const int h2 = h_ZSH[l2], k2 = h_ZSK[l2];
        const bool down = (i == 6 || i == 12);
        const int Hin  = (i <= 6) ? 32 : (i <= 12) ? 16 : 8;
        const int Cin1 = k1 * 16, Cout1 = h1 * 16;
        const int Cin2 = k2 * 16, Cout2 = h2 * 16;

        if (down) {
            const float* rw = (i == 6) ? WR6 : WR12;
            const float* rb = (i == 6) ? res_b6 : res_b12;
            launch_conv(A, rw, rb, bng, bnb, bnm, bnv, /*bnrow=*/-1,
                        nullptr, R, Bn, Cin1, Cout1, Hin, Hin,
                        /*stride=*/2, /*ksize=*/1, /*relu=*/0, stream);
        }

        // conv w1 -> Bb : BN(1+2i) + ReLU
        launch_conv(A, WKH + (long)h_OFFS[l1] * 2304, nullptr,
                    bng, bnb, bnm, bnv, 1 + 2 * i, nullptr, Bb,
                    Bn, Cin1, Cout1, Hin, Hin,
                    down ? 2 : 1, 3, 1, stream);

        const int Hmid = down ? Hin / 2 : Hin;

        // conv w2 -> A : BN(2+2i) + residual + ReLU
        launch_conv(Bb, WKH + (long)h_OFFS[l2] * 2304, nullptr,
                    bng, bnb, bnm, bnv, 2 + 2 * i,
                    down ? (const float*)R : (const float*)A, A,
                    Bn, Cin2, Cout2, Hmid, Hmid,
                    1, 3, 1, stream);
    }

    // 5) global average pool + final linear -> out
    pool_fc<<<Bn, 64, 0, stream>>>(A, fin_w, fin_b, out);
}